// Attention_51350628991083
// MI455X (gfx1250) — compile-verified
//
#include <hip/hip_runtime.h>
#include <math.h>

#define B_ 64
#define L_ 4096
#define H_ 512
#define D_ 1024
#define SPLIT 8

typedef __attribute__((ext_vector_type(2))) float v2f;
typedef __attribute__((ext_vector_type(8))) float v8f;

// ---------------------------------------------------------------------------
// Kernel 1: q[b,o] = sum_h query[b,h] * W_q[o,h]   (64x512x512 fp32 GEMM)
// One wave32 per 16x16 output tile, V_WMMA_F32_16X16X4_F32 over K.
// A layout (16x4 f32): lanes 0-15 hold M=lane {K=k,k+1}; lanes 16-31 {K=k+2,k+3}
// B layout (4x16 f32): symmetric, N over lanes, K pairs over lane halves.
// C layout: VGPR r -> row r (lanes 0-15) / row r+8 (lanes 16-31), col = lane%16.
// ---------------------------------------------------------------------------
__global__ __launch_bounds__(32) void qproj_wmma(const float* __restrict__ query,
                                                 const float* __restrict__ Wq,
                                                 float* __restrict__ qout) {
    const int nt   = blockIdx.x & 31;   // 32 tiles along N (=H out)
    const int mt   = blockIdx.x >> 5;   // 4 tiles along M (=B)
    const int m0   = mt * 16, n0 = nt * 16;
    const int lane = threadIdx.x;
    const int half = lane >> 4;         // which K-pair this lane holds
    const int lm   = lane & 15;

    const float* arow = query + (size_t)(m0 + lm) * H_ + 2 * half;
    const float* brow = Wq    + (size_t)(n0 + lm) * H_ + 2 * half;   // W_q[o,h]: B is W_q^T

    v8f c = {};
    for (int k = 0; k < H_; k += 4) {
        v2f a = *(const v2f*)(arow + k);
        v2f b = *(const v2f*)(brow + k);
        c = __builtin_amdgcn_wmma_f32_16x16x4_f32(false, a, false, b,
                                                  (short)0, c, false, false);
    }
#pragma unroll
    for (int r = 0; r < 8; ++r)
        qout[(size_t)(m0 + r + 8 * half) * H_ + n0 + lm] = c[r];
}

// ---------------------------------------------------------------------------
// Kernel 2: scores[b,l] = sum_h v_e[h] * tanh(q[b,h] + proj_key[b,l,h])
// Block = 256 threads = 8 waves; block handles 32 rows of one batch b.
// q_b and v_e staged in LDS once per block; proj_key streamed with float4
// fully-coalesced loads (lane j reads bytes [16j,16j+16) of each 512B chunk).
// ---------------------------------------------------------------------------
__global__ __launch_bounds__(256) void scores_kernel(const float* __restrict__ qproj,
                                                     const float* __restrict__ key,
                                                     const float* __restrict__ ve,
                                                     const int*   __restrict__ mask,
                                                     float* __restrict__ scores) {
    __shared__ float sq[H_];
    __shared__ float sv[H_];
    const int ROWS = 32;
    const int b  = blockIdx.x / (L_ / ROWS);
    const int lc = blockIdx.x % (L_ / ROWS);
    const int t  = threadIdx.x;

    sq[t]       = qproj[(size_t)b * H_ + t];
    sq[t + 256] = qproj[(size_t)b * H_ + t + 256];
    sv[t]       = ve[t];
    sv[t + 256] = ve[t + 256];
    __syncthreads();

    const int wave = t >> 5, lane = t & 31;
#pragma unroll
    for (int r = 0; r < 4; ++r) {
        const int l = lc * ROWS + wave * 4 + r;
        const float* kp = key + ((size_t)b * L_ + l) * H_;
        float acc = 0.f;
#pragma unroll
        for (int i = 0; i < 4; ++i) {
            const int idx = i * 128 + lane * 4;
            float4 fk = *(const float4*)(kp + idx);
            acc += sv[idx + 0] * tanhf(sq[idx + 0] + fk.x);
            acc += sv[idx + 1] * tanhf(sq[idx + 1] + fk.y);
            acc += sv[idx + 2] * tanhf(sq[idx + 2] + fk.z);
            acc += sv[idx + 3] * tanhf(sq[idx + 3] + fk.w);
        }
#pragma unroll
        for (int off = 16; off > 0; off >>= 1)
            acc += __shfl_down(acc, off, 32);
        if (lane == 0) {
            float s = (mask[(size_t)b * L_ + l] == 0) ? -INFINITY : acc;
            scores[(size_t)b * L_ + l] = s;
        }
    }
}

// ---------------------------------------------------------------------------
// Kernel 3: in-place masked softmax over L per batch row (scores -> alphas).
// One block per b; 16 elements per thread held in registers between passes.
// ---------------------------------------------------------------------------
__global__ __launch_bounds__(256) void softmax_kernel(float* __restrict__ alphas) {
    __shared__ float red[256];
    const int b = blockIdx.x;
    const int t = threadIdx.x;
    float loc[16];

    float m = -INFINITY;
#pragma unroll
    for (int i = 0; i < 16; ++i) {
        loc[i] = alphas[(size_t)b * L_ + i * 256 + t];
        m = fmaxf(m, loc[i]);
    }
    red[t] = m; __syncthreads();
    for (int s = 128; s > 0; s >>= 1) {
        if (t < s) red[t] = fmaxf(red[t], red[t + s]);
        __syncthreads();
    }
    m = red[0]; __syncthreads();

    float sum = 0.f;
#pragma unroll
    for (int i = 0; i < 16; ++i) { loc[i] = __expf(loc[i] - m); sum += loc[i]; }
    red[t] = sum; __syncthreads();
    for (int s = 128; s > 0; s >>= 1) {
        if (t < s) red[t] += red[t + s];
        __syncthreads();
    }
    const float inv = 1.f / red[0];
#pragma unroll
    for (int i = 0; i < 16; ++i)
        alphas[(size_t)b * L_ + i * 256 + t] = loc[i] * inv;
}

// ---------------------------------------------------------------------------
// Kernel 4: partial context sums; split L 8 ways for memory-channel
// parallelism (2048 blocks streaming the 1GB value tensor, coalesced).
// alpha loads are block-uniform -> scalar path; fp order is fixed ->
// bit-deterministic across replays (no atomics).
// ---------------------------------------------------------------------------
__global__ __launch_bounds__(256) void context_partial(const float* __restrict__ alphas,
                                                       const float* __restrict__ value,
                                                       float* __restrict__ part) {
    const int t      = threadIdx.x;
    const int dchunk = blockIdx.x & 3;          // D/256 = 4
    const int s      = (blockIdx.x >> 2) & (SPLIT - 1);
    const int b      = blockIdx.x >> 5;
    const int d      = dchunk * 256 + t;
    const int LCH    = L_ / SPLIT;              // 512

    const float* ap = alphas + (size_t)b * L_ + s * LCH;
    const float* vp = value + ((size_t)b * L_ + (size_t)s * LCH) * D_ + d;

    float acc = 0.f;
    for (int l = 0; l < LCH; l += 4) {
        __builtin_prefetch(vp + (size_t)(l + 16) * D_, 0, 1);  // global_prefetch_b8
        acc += ap[l + 0] * vp[(size_t)(l + 0) * D_];
        acc += ap[l + 1] * vp[(size_t)(l + 1) * D_];
        acc += ap[l + 2] * vp[(size_t)(l + 2) * D_];
        acc += ap[l + 3] * vp[(size_t)(l + 3) * D_];
    }
    part[(size_t)(b * SPLIT + s) * D_ + d] = acc;
}

// Kernel 5: deterministic sequential (s = 0..SPLIT-1) reduction of the SPLIT partials -> context.
__global__ __launch_bounds__(256) void context_reduce(const float* __restrict__ part,
                                                      float* __restrict__ ctx) {
    const int idx = blockIdx.x * 256 + threadIdx.x;   // over B*D = 65536
    const int b = idx >> 10, d = idx & (D_ - 1);
    float acc = 0.f;
#pragma unroll
    for (int s = 0; s < SPLIT; ++s)
        acc += part[(size_t)(b * SPLIT + s) * D_ + d];
    ctx[idx] = acc;
}

extern "C" void kernel_launch(void* const* d_in, const int* in_sizes, int n_in,
                              void* d_out, int out_size, void* d_ws, size_t ws_size,
                              hipStream_t stream) {
    const float* query = (const float*)d_in[0];   // [B,1,H]
    const float* key   = (const float*)d_in[1];   // [B,L,H]
    const float* value = (const float*)d_in[2];   // [B,L,D]
    const int*   mask  = (const int*)  d_in[3];   // [B,L]
    const float* Wq    = (const float*)d_in[4];   // [H,H]
    const float* ve    = (const float*)d_in[5];   // [H]

    float* ctx    = (float*)d_out;                // [B,1,D] first output
    float* alphas = (float*)d_out + B_ * D_;      // [B,1,L] second output
    float* qproj  = (float*)d_ws;                 // [B,H]        (128 KB)
    float* part   = (float*)d_ws + B_ * H_;       // [B,SPLIT,D]  (2 MB)

    qproj_wmma     <<<(B_ / 16) * (H_ / 16),       32, 0, stream>>>(query, Wq, qproj);
    scores_kernel  <<<B_ * (L_ / 32),             256, 0, stream>>>(qproj, key, ve, mask, alphas);
    softmax_kernel <<<B_,                         256, 0, stream>>>(alphas);
    context_partial<<<B_ * SPLIT * (D_ / 256),    256, 0, stream>>>(alphas, value, part);
    context_reduce <<<(B_ * D_) / 256,            256, 0, stream>>>(part, ctx);
}